// PatchGATAggregationBlock_52510270161515
// MI455X (gfx1250) — compile-verified
//
#include <hip/hip_runtime.h>
#include <hip/hip_bf16.h>

typedef __attribute__((ext_vector_type(2))) float v2f;
typedef __attribute__((ext_vector_type(8))) float v8f;
typedef __attribute__((ext_vector_type(4))) int   v4i;

#define HIDDEN 128

#if defined(__HIP_DEVICE_COMPILE__) && \
    __has_builtin(__builtin_amdgcn_global_load_async_to_lds_b128) && \
    __has_builtin(__builtin_amdgcn_s_wait_asynccnt)
#define HAVE_ASYNC_LDS 1
typedef __attribute__((address_space(1))) v4i GLOB_v4i;
typedef __attribute__((address_space(3))) v4i LDS_v4i;
#else
#define HAVE_ASYNC_LDS 0
#endif

__device__ __forceinline__ float lrelu(float x) { return x > 0.0f ? x : 0.2f * x; }

// --------------------------------------------------------------------------
// u_s = W^T a_src, u_d = W^T a_dst  per layer (tiny precompute)
// U layout: [layer][ us(128) | ud(128) ]
// --------------------------------------------------------------------------
__global__ void prep_u_kernel(const float* __restrict__ W0, const float* __restrict__ s0, const float* __restrict__ d0,
                              const float* __restrict__ W1, const float* __restrict__ s1, const float* __restrict__ d1,
                              const float* __restrict__ W2, const float* __restrict__ s2, const float* __restrict__ d2,
                              float* __restrict__ U) {
  const int l = blockIdx.x;
  const int k = threadIdx.x;           // 0..127
  const float* W  = (l == 0) ? W0 : (l == 1) ? W1 : W2;
  const float* as = (l == 0) ? s0 : (l == 1) ? s1 : s2;
  const float* ad = (l == 0) ? d0 : (l == 1) ? d1 : d2;
  float us = 0.0f, ud = 0.0f;
  for (int e = 0; e < HIDDEN; ++e) {
    float we = W[e * HIDDEN + k];
    us = fmaf(as[e], we, us);
    ud = fmaf(ad[e], we, ud);
  }
  U[l * 2 * HIDDEN + k] = us;
  U[l * 2 * HIDDEN + HIDDEN + k] = ud;
}

// --------------------------------------------------------------------------
// Fused GAT-on-chain + masked-mean pooling, algebraically collapsed:
//   s[n] = x[n]·u_s ; t[n] = x[n]·u_d
//   3-edge softmax per dst; w[src] = sum_dst mask[dst]*alpha(src->dst)
//   Y[g] = (sum_n w[n] x[n]) / max(cnt,1) ;  flag[g] = (cnt>0)
// One block (256 threads) per graph. X tile staged in LDS via CDNA5
// GLOBAL_LOAD_ASYNC_TO_LDS_B128 (ASYNCcnt) when available.
// All LDS phases sweep the contiguous n dimension across lanes -> no bank
// conflicts, no padding, 16B-aligned rows for the B128 async path.
// --------------------------------------------------------------------------
template <int PL>
__global__ void gat_pool_kernel(const float* __restrict__ xin, const int* __restrict__ lengths,
                                const float* __restrict__ Ul,
                                float* __restrict__ Yg, float* __restrict__ flag,
                                int P, int T) {
  extern __shared__ float sm[];
  float* Xs  = sm;                            // [HIDDEN][PL], row stride PL
  float* s_  = Xs + HIDDEN * PL;              // PL
  float* t_  = s_ + PL;                       // PL
  float* ap  = t_ + PL;                       // PL  alpha from prev (masked)
  float* asf = ap + PL;                       // PL  alpha self      (masked)
  float* an  = asf + PL;                      // PL  alpha from next (masked)
  float* wg  = an + PL;                       // PL  node weights
  float* uu  = wg + PL;                       // 2*HIDDEN: us | ud
  float* red = uu + 2 * HIDDEN;               // HIDDEN (y[k] partials)
  float* cnt = red + HIDDEN;                  // 1

  const int tid  = threadIdx.x;
  const int lane = tid & 31;
  const int wv   = tid >> 5;
  const int g = blockIdx.x;
  const int b = g / P, p = g - b * P;

  if (tid < 2 * HIDDEN) uu[tid] = Ul[tid];
  if (tid == 0) cnt[0] = 0.0f;

  // ---- Stage X tile: Xs[k*PL + n] = x[b, k, p*PL + n] ----
  const size_t base = (size_t)b * HIDDEN * T + (size_t)p * PL;
#if HAVE_ASYNC_LDS
  if constexpr ((PL & 3) == 0) {
    constexpr int NV = HIDDEN * PL / 4;       // float4 transfers
    constexpr int RV = PL / 4;                // vec4 per row
    for (int idx = tid; idx < NV; idx += 256) {
      const int k = idx / RV;
      const int n = (idx - k * RV) * 4;
      __builtin_amdgcn_global_load_async_to_lds_b128(
          (GLOB_v4i*)(uintptr_t)(xin + base + (size_t)k * T + n),
          (LDS_v4i*)(uint32_t)(uintptr_t)(Xs + idx * 4), 0, 0);
    }
    __builtin_amdgcn_s_wait_asynccnt(0);
  } else
#endif
  {
    for (int idx = tid; idx < HIDDEN * PL; idx += 256) {
      const int k = idx / PL, n = idx - k * PL;
      Xs[idx] = xin[base + (size_t)k * T + n];
    }
  }
  __syncthreads();

  // ---- s, t per node (lanes over n: conflict-free) ----
  if (tid < PL) {
    float ss = 0.0f, tt = 0.0f;
    for (int k = 0; k < HIDDEN; ++k) {
      float xv = Xs[k * PL + tid];
      ss = fmaf(xv, uu[k], ss);
      tt = fmaf(xv, uu[HIDDEN + k], tt);
    }
    s_[tid] = ss;
    t_[tid] = tt;
  }
  __syncthreads();

  // ---- softmax over incoming edges of dst = tid; fold mask[dst] in ----
  if (tid < PL) {
    float td = t_[tid];
    float ls = lrelu(s_[tid] + td);
    float lp = (tid > 0)      ? lrelu(s_[tid - 1] + td) : -3.0e38f;
    float ln = (tid < PL - 1) ? lrelu(s_[tid + 1] + td) : -3.0e38f;
    float mx = fmaxf(ls, fmaxf(lp, ln));
    float ep = (tid > 0)      ? __expf(lp - mx) : 0.0f;
    float es = __expf(ls - mx);
    float en = (tid < PL - 1) ? __expf(ln - mx) : 0.0f;
    float inv = 1.0f / (ep + es + en);
    float m = 1.0f;
    if (lengths) m = ((p * PL + tid) < lengths[b]) ? 1.0f : 0.0f;
    ap[tid]  = m * ep * inv;
    asf[tid] = m * es * inv;
    an[tid]  = m * en * inv;
    atomicAdd(cnt, m);
  }
  __syncthreads();

  // ---- node weights: w[n] = asf[n] + ap[n+1] + an[n-1] ----
  if (tid < PL) {
    float w = asf[tid];
    if (tid + 1 < PL) w += ap[tid + 1];
    if (tid > 0)      w += an[tid - 1];
    wg[tid] = w;
  }
  __syncthreads();

  // ---- y[k] = sum_n w[n] * Xs[k][n]: wave owns k, lanes sweep n ----
  for (int j = 0; j < HIDDEN / 8; ++j) {      // 8 waves x 16 k-values
    const int k = wv * (HIDDEN / 8) + j;
    float acc = 0.0f;
    for (int n = lane; n < PL; n += 32) acc = fmaf(wg[n], Xs[k * PL + n], acc);
    for (int off = 16; off > 0; off >>= 1) acc += __shfl_down(acc, off, 32);
    if (lane == 0) red[k] = acc;
  }
  __syncthreads();

  const float c = cnt[0];
  if (tid < HIDDEN) {
    Yg[(size_t)g * HIDDEN + tid] = red[tid] * (1.0f / fmaxf(c, 1.0f));
  }
  if (tid == 0) flag[g] = (c > 0.0f) ? 1.0f : 0.0f;
}

// --------------------------------------------------------------------------
// feat = Y @ W^T + flag*b, scattered to [B, d, P] layout.
// One wave per 16x16 tile, V_WMMA_F32_16X16X4_F32, K = 128 in steps of 4.
// block = 256 threads (8 waves -> full N=128); grid.x = G/16 M-tiles.
// --------------------------------------------------------------------------
__global__ void proj_wmma_kernel(const float* __restrict__ Yg, const float* __restrict__ Wm,
                                 const float* __restrict__ bias, const float* __restrict__ flag,
                                 float* __restrict__ xout, int P) {
  const int lane = threadIdx.x & 31;
  const int wave = threadIdx.x >> 5;          // N tile 0..7
  const int m0 = blockIdx.x * 16;
  const int n0 = wave * 16;
  const int lh  = lane >> 4;                  // lane half selects K pair {0,1} vs {2,3}
  const int l15 = lane & 15;

  // A (16x4 f32): lane half h holds K = k0 + 2h + {0,1} for row m0+l15
  const float* arow = Yg + (size_t)(m0 + l15) * HIDDEN + lh * 2;
  // B (4x16 f32): B[k][n] = W[n][k]; lane half h holds K = k0 + 2h + {0,1} for col n0+l15
  const float* brow = Wm + (size_t)(n0 + l15) * HIDDEN + lh * 2;

  v8f c = {};
  for (int k0 = 0; k0 < HIDDEN; k0 += 4) {
    v2f a, bb;
    a.x  = arow[k0];
    a.y  = arow[k0 + 1];
    bb.x = brow[k0];
    bb.y = brow[k0 + 1];
    c = __builtin_amdgcn_wmma_f32_16x16x4_f32(false, a, false, bb, (short)0, c, false, false);
  }

  const int e = n0 + l15;
  const float be = bias[e];
  for (int r = 0; r < 8; ++r) {
    const int gm = m0 + lh * 8 + r;           // C/D layout: VGPR r -> M=r (lanes 0-15) / M=8+r
    const float val = c[r] + flag[gm] * be;
    const int gb = gm / P, gp = gm - gb * P;
    xout[((size_t)gb * HIDDEN + e) * P + gp] = val;
  }
}

// --------------------------------------------------------------------------
extern "C" void kernel_launch(void* const* d_in, const int* in_sizes, int n_in,
                              void* d_out, int out_size, void* d_ws, size_t ws_size,
                              hipStream_t stream) {
  (void)in_sizes; (void)n_in; (void)out_size; (void)ws_size;

  const float* x   = (const float*)d_in[0];
  const int* lens  = (const int*)d_in[1];
  const float* W0  = (const float*)d_in[2];
  const float* as0 = (const float*)d_in[3];
  const float* ad0 = (const float*)d_in[4];
  const float* b0  = (const float*)d_in[5];
  const float* W1  = (const float*)d_in[6];
  const float* as1 = (const float*)d_in[7];
  const float* ad1 = (const float*)d_in[8];
  const float* b1  = (const float*)d_in[9];
  const float* W2  = (const float*)d_in[10];
  const float* as2 = (const float*)d_in[11];
  const float* ad2 = (const float*)d_in[12];
  const float* b2  = (const float*)d_in[13];

  float* ws   = (float*)d_ws;
  float* U    = ws;                 // 3*256
  float* flag = ws + 768;           // up to 1024
  float* Y    = ws + 1792;          // up to 1024*128
  float* x1   = Y + 131072;         // [64,128,16]
  float* x2   = x1 + 131072;        // [64,128,4]

  // pl=256 tile needs ~136KB dynamic LDS (CDNA5 WGP has 320KB)
  (void)hipFuncSetAttribute(reinterpret_cast<const void*>(gat_pool_kernel<256>),
                            hipFuncAttributeMaxDynamicSharedMemorySize, 150 * 1024);
  (void)hipFuncSetAttribute(reinterpret_cast<const void*>(gat_pool_kernel<4>),
                            hipFuncAttributeMaxDynamicSharedMemorySize, 150 * 1024);
  (void)hipFuncSetAttribute(reinterpret_cast<const void*>(gat_pool_kernel<1>),
                            hipFuncAttributeMaxDynamicSharedMemorySize, 150 * 1024);

  auto smem = [](int pl) -> size_t {
    return (size_t)(HIDDEN * pl + 6 * pl + 2 * HIDDEN + HIDDEN + 8) * sizeof(float);
  };

  prep_u_kernel<<<3, 128, 0, stream>>>(W0, as0, ad0, W1, as1, ad1, W2, as2, ad2, U);

  // Layer 0: pl=256, P=16, T=4096, G=1024 (uses lengths mask)
  gat_pool_kernel<256><<<1024, 256, smem(256), stream>>>(x, lens, U, Y, flag, 16, 4096);
  proj_wmma_kernel<<<1024 / 16, 256, 0, stream>>>(Y, W0, b0, flag, x1, 16);

  // Layer 1: pl=4, P=4, T=16, G=256 (mask all-true)
  gat_pool_kernel<4><<<256, 256, smem(4), stream>>>(x1, nullptr, U + 256, Y, flag, 4, 16);
  proj_wmma_kernel<<<256 / 16, 256, 0, stream>>>(Y, W1, b1, flag, x2, 4);

  // Layer 2: pl=1, P=4, T=4, G=256 (self-loop only)
  gat_pool_kernel<1><<<256, 256, smem(1), stream>>>(x2, nullptr, U + 512, Y, flag, 4, 4);
  proj_wmma_kernel<<<256 / 16, 256, 0, stream>>>(Y, W2, b2, flag, (float*)d_out, 4);
}